// GNNHist_84636625535683
// MI455X (gfx1250) — compile-verified
//
#include <hip/hip_runtime.h>
#include <hip/hip_bf16.h>
#include <math.h>

// Problem constants (fixed by the reference harness)
#define BB    2048
#define U1    101
#define DD    128
#define STEP  165
#define II    64          // i
#define IM1   63          // i-1
#define NH    200         // hidden width
#define MTOT  (BB * U1)   // 206848 = 12928 * 16 exactly
#define MT    (MTOT / 16) // 12928 M-tiles
#define NT    13          // ceil(200/16)

typedef __attribute__((ext_vector_type(2))) float v2f;
typedef __attribute__((ext_vector_type(8))) float v8f;

// ---------------------------------------------------------------------------
// Kernel 1: per-batch-row preprocessing.
// Computes mean_emb, incoming, hist-mean, gathered-selected-mean, then
// step_context = [sel_mean|hist_mean] @ W_sc + b_sc, then folds all
// u-invariant feature blocks of s@W1 into base[b, 0:200]:
//   base = b1 + 0.5*W1[1,:] + incoming@W1[2:130] + ctx@W1[258:386] + mean@W1[386:514]
// ---------------------------------------------------------------------------
__global__ __launch_bounds__(256) void k1_prep(
    const float* __restrict__ emb, const float* __restrict__ Wsc,
    const float* __restrict__ bsc, const float* __restrict__ W1,
    const float* __restrict__ b1,  const int* __restrict__ seq,
    float* __restrict__ base)
{
    __shared__ float s_mean[DD], s_sel[DD], s_hist[DD], s_inc[DD], s_ctx[DD];
    __shared__ int   s_seq[IM1];
    const int b = blockIdx.x;
    const int t = threadIdx.x;
    const float* eb = emb + (size_t)b * (STEP * DD);

    if (t < IM1) s_seq[t] = seq[b * IM1 + t];
    __syncthreads();

    if (t < DD) {
        float tot = 0.f, hist = 0.f, inc = 0.f;
        for (int s = 0; s < STEP; ++s) {
            float v = eb[s * DD + t];
            tot += v;
            if (s >= U1 && s < U1 + IM1) hist += v;   // rows 101..163
            if (s == STEP - 1) inc = v;               // row 164
        }
        float sel = 0.f;
        for (int j = 0; j < IM1; ++j) {
            // embeddings.reshape(-1,D)[b*63 + seq[b,j]]
            size_t ridx = (size_t)(b * IM1 + s_seq[j]) * DD;
            sel += emb[ridx + t];
        }
        s_mean[t] = tot  * (1.0f / (float)STEP);
        s_hist[t] = hist * (1.0f / (float)IM1);
        s_sel[t]  = sel  * (1.0f / (float)IM1);
        s_inc[t]  = inc;
    }
    __syncthreads();

    if (t < DD) {  // step_context = [sel|hist] @ W_sc + b_sc
        float acc = bsc[t];
        for (int k = 0; k < DD; ++k) acc += s_sel[k]  * Wsc[k * DD + t];
        for (int k = 0; k < DD; ++k) acc += s_hist[k] * Wsc[(DD + k) * DD + t];
        s_ctx[t] = acc;
    }
    __syncthreads();

    if (t < NH) {
        float acc = b1[t] + 0.5f * W1[NH + t];   // idx_col = i/v_size = 0.5
        for (int k = 0; k < DD; ++k) acc += s_inc[k]  * W1[(2 + k) * NH + t];
        for (int k = 0; k < DD; ++k) acc += s_ctx[k]  * W1[(258 + k) * NH + t];
        for (int k = 0; k < DD; ++k) acc += s_mean[k] * W1[(386 + k) * NH + t];
        base[(size_t)b * NH + t] = acc;
    }
}

// ---------------------------------------------------------------------------
// Kernel 2: fused WMMA GEMM + epilogue.
// pi[m] = relu( emb[b,u,:] @ W1[130:258] + base[b,:] + w[m]*W1[0,:] ) @ W2 + b2
// One wave per 16-row M tile, exact fp32 via V_WMMA_F32_16X16X4_F32 (K=4,
// 32 MMAs per 16x16x128 tile). B tile staged in LDS per N-tile (13 tiles).
// ---------------------------------------------------------------------------
__global__ __launch_bounds__(256) void k2_gemm(
    const float* __restrict__ emb, const float* __restrict__ wv,
    const float* __restrict__ W1,  const float* __restrict__ W2,
    const float* __restrict__ b2,  const float* __restrict__ base,
    float* __restrict__ pi)
{
    __shared__ float sB[DD * 16];     // current B tile: [k][ncol], k=0..127
    __shared__ float s_w2[208];       // W2 padded (cols 200..207 = 0)
    __shared__ float s_w10[208];      // W1 row 0 padded

    const int t     = threadIdx.x;
    const int wave  = t >> 5;
    const int lane  = t & 31;
    const int mrow  = lane & 15;      // A: lane holds row M = lane%16
    const int khalf = lane >> 4;      // A/B: lanes 16-31 hold K=2,3 of each 4
    const int ncol  = lane & 15;      // C/D: lane holds column N = lane%16
    const int mt    = blockIdx.x * 8 + wave;   // M-tile, < 12928

    if (t < 208) {
        s_w2[t]  = (t < NH) ? W2[t] : 0.f;
        s_w10[t] = (t < NH) ? W1[t] : 0.f;
    }

    // ---- Preload A fragments (kept in registers across all 13 N tiles) ----
    const int mg = mt * 16 + mrow;            // this lane's A row
    const int bA = mg / U1, uA = mg % U1;
    const float* arow = emb + ((size_t)bA * STEP + uA) * DD;
    v2f A[32];
    #pragma unroll
    for (int kk = 0; kk < 32; ++kk) {
        const float2 av = *reinterpret_cast<const float2*>(arow + kk * 4 + khalf * 2);
        A[kk].x = av.x; A[kk].y = av.y;
    }

    // ---- Per-lane epilogue metadata: this lane's 8 output rows ----
    int   boff[8];   // base row offsets
    float wr[8];     // w[b,u] per row
    #pragma unroll
    for (int r = 0; r < 8; ++r) {
        int m2  = mt * 16 + khalf * 8 + r;    // C/D: VGPR r = row khalf*8+r
        boff[r] = (m2 / U1) * NH;
        wr[r]   = wv[m2];
    }

    float pip[8];
    #pragma unroll
    for (int r = 0; r < 8; ++r) pip[r] = 0.f;

    for (int nt = 0; nt < NT; ++nt) {
        __syncthreads();
        // Cooperative load of B tile: W1 rows 130..257, cols nt*16..nt*16+15
        #pragma unroll
        for (int it = 0; it < 8; ++it) {
            int idx = it * 256 + t;           // 0..2047
            int k = idx >> 4, nc = idx & 15;
            int n = nt * 16 + nc;
            sB[idx] = (n < NH) ? W1[(130 + k) * NH + n] : 0.f;
        }
        __syncthreads();

        v8f acc = {0.f, 0.f, 0.f, 0.f, 0.f, 0.f, 0.f, 0.f};
        #pragma unroll
        for (int kk = 0; kk < 32; ++kk) {
            const int k0 = kk * 4 + khalf * 2;    // B: lanes16-31 hold K=2,3
            v2f bf;
            bf.x = sB[k0 * 16 + ncol];
            bf.y = sB[(k0 + 1) * 16 + ncol];
            acc = __builtin_amdgcn_wmma_f32_16x16x4_f32(
                    false, A[kk], false, bf, (short)0, acc, false, false);
        }

        // Fused epilogue for this N tile: column n = nt*16 + ncol
        const int   n   = nt * 16 + ncol;
        const float wc  = s_w2[n];                 // 0 for padded cols
        const float w10 = s_w10[n];
        #pragma unroll
        for (int r = 0; r < 8; ++r) {
            float h = acc[r] + base[boff[r] + n] + wr[r] * w10;
            pip[r] += fmaxf(h, 0.f) * wc;          // wc=0 kills n>=200 garbage
        }
    }

    // Reduce over the 16 N-columns (lanes 0-15 and 16-31 independently)
    const float b2v = b2[0];
    #pragma unroll
    for (int r = 0; r < 8; ++r) {
        float v = pip[r];
        v += __shfl_xor(v, 1, 32);
        v += __shfl_xor(v, 2, 32);
        v += __shfl_xor(v, 4, 32);
        v += __shfl_xor(v, 8, 32);
        pip[r] = v;
    }
    if ((lane & 15) == 0) {   // lane 0 -> rows 0..7, lane 16 -> rows 8..15
        #pragma unroll
        for (int r = 0; r < 8; ++r) {
            int m2 = mt * 16 + khalf * 8 + r;
            pi[m2] = pip[r] + b2v;
        }
    }
}

// ---------------------------------------------------------------------------
// Kernel 3: mask, log_softmax over 101, first-occurrence argmax.
// out[0..B) = selected (as float), out[B + b*101 + u] = log-prob.
// mask is a jax bool array -> 1 byte per element.
// ---------------------------------------------------------------------------
__global__ __launch_bounds__(128) void k3_softmax(
    const float* __restrict__ pi, const unsigned char* __restrict__ mask,
    float* __restrict__ out)
{
    __shared__ float sv[U1];
    __shared__ float s_max, s_lse;
    __shared__ int   s_arg;
    const int b = blockIdx.x, t = threadIdx.x;

    if (t < U1)
        sv[t] = mask[b * U1 + t] ? -1000000.0f : pi[b * U1 + t];
    __syncthreads();

    if (t == 0) {
        float m = sv[0]; int arg = 0;
        for (int j = 1; j < U1; ++j)
            if (sv[j] > m) { m = sv[j]; arg = j; }
        float s = 0.f;
        for (int j = 0; j < U1; ++j) s += expf(sv[j] - m);
        s_max = m; s_lse = logf(s); s_arg = arg;
    }
    __syncthreads();

    if (t < U1) out[BB + b * U1 + t] = sv[t] - s_max - s_lse;
    if (t == 0) out[b] = (float)s_arg;
}

// ---------------------------------------------------------------------------
extern "C" void kernel_launch(void* const* d_in, const int* in_sizes, int n_in,
                              void* d_out, int out_size, void* d_ws, size_t ws_size,
                              hipStream_t stream)
{
    const float* emb  = (const float*)d_in[0];
    const float* w    = (const float*)d_in[1];
    const float* Wsc  = (const float*)d_in[2];
    const float* bsc  = (const float*)d_in[3];
    const float* W1   = (const float*)d_in[4];
    const float* b1   = (const float*)d_in[5];
    const float* W2   = (const float*)d_in[6];
    const float* b2   = (const float*)d_in[7];
    const int*   seq  = (const int*)d_in[8];
    const unsigned char* mask = (const unsigned char*)d_in[9];
    // d_in[10] = i, d_in[11] = v_size : compile-time constants here.

    float* base = (float*)d_ws;            // B*200 floats
    float* pi   = base + (size_t)BB * NH;  // B*101 floats
    float* out  = (float*)d_out;

    k1_prep   <<<BB, 256, 0, stream>>>(emb, Wsc, bsc, W1, b1, seq, base);
    k2_gemm   <<<MT / 8, 256, 0, stream>>>(emb, w, W1, W2, b2, base, pi);
    k3_softmax<<<BB, 128, 0, stream>>>(pi, mask, out);
}